// MLPAttention_15676630630946
// MI455X (gfx1250) — compile-verified
//
#include <hip/hip_runtime.h>
#include <hip/hip_bf16.h>
#include <math.h>

// Problem constants (from reference)
#define B_   4
#define DL_  256
#define EL_  512
#define DDEC 512   // == DENC == K for both projections
#define H_   256   // N for both projections

typedef __attribute__((ext_vector_type(2))) float v2f;
typedef __attribute__((ext_vector_type(8))) float v8f;

#if __has_builtin(__builtin_amdgcn_tanhf)
#define TANH(x) __builtin_amdgcn_tanhf(x)
#elif __has_builtin(__builtin_amdgcn_tanh_f32)
#define TANH(x) __builtin_amdgcn_tanh_f32(x)
#else
#define TANH(x) tanhf(x)
#endif

// ---------------------------------------------------------------------------
// Phase 1: fused projection GEMMs via V_WMMA_F32_16X16X4_F32.
// One wave computes one 16x16 tile of either dec_f (tiles [0,1024)) or
// enc_f (tiles [1024,3072)).  K = 512 -> 128 WMMA steps per tile.
//
// fp32 WMMA operand layouts (CDNA5 ISA 7.12.2):
//   A 16x4 : lane = M + 16*(K>=2), vgpr k&1          -> contiguous b64 load/lane
//   B 4x16 : lane = N + 16*(K>=2), vgpr k&1 (mirror of A)
//   C 16x16: vgpr r holds M=r (lanes 0-15) / M=8+r (lanes 16-31), N = lane&15
// ---------------------------------------------------------------------------
__global__ void __launch_bounds__(256)
mlp_attn_proj_wmma(const float* __restrict__ dec_in,
                   const float* __restrict__ enc_in,
                   const float* __restrict__ Wq,
                   const float* __restrict__ bq,
                   const float* __restrict__ Wc,
                   float* __restrict__ dec_f,
                   float* __restrict__ enc_f) {
  const int wave = (blockIdx.x * blockDim.x + threadIdx.x) >> 5;   // 0..3071
  const int lane = threadIdx.x & 31;

  const bool is_dec = (wave < (DL_ * B_ / 16) * (H_ / 16));        // < 1024
  const int  t      = is_dec ? wave : wave - 1024;
  const int  mt     = t >> 4;          // row-tile
  const int  nt     = t & 15;          // col-tile

  const float* __restrict__ A = is_dec ? dec_in : enc_in;
  const float* __restrict__ W = is_dec ? Wq : Wc;
  float* __restrict__ Out     = is_dec ? dec_f : enc_f;

  const int m      = lane & 15;        // M for A loads, N for B loads
  const int khalf  = (lane >> 4) << 1; // 0 or 2
  const int arow   = mt * 16 + m;      // A row this lane feeds
  const int bcol   = nt * 16 + m;      // W column this lane feeds

  const float* __restrict__ ap = A + (size_t)arow * DDEC;

  v8f acc = {};
#pragma unroll 4
  for (int k = 0; k < DDEC; k += 4) {
    v2f a, b;
    a.x = ap[k + khalf + 0];                       // A[arow, k+khalf+0]
    a.y = ap[k + khalf + 1];                       // A[arow, k+khalf+1]  (b64)
    b.x = W[(size_t)(k + khalf + 0) * H_ + bcol];  // W[k+khalf+0, bcol]
    b.y = W[(size_t)(k + khalf + 1) * H_ + bcol];  // W[k+khalf+1, bcol]
    acc = __builtin_amdgcn_wmma_f32_16x16x4_f32(
        /*neg_a=*/false, a, /*neg_b=*/false, b,
        /*c_mod=*/(short)0, acc, /*reuse_a=*/false, /*reuse_b=*/false);
  }

  // Epilogue: scatter C tile, fusing +bq for the decoder projection.
  const int n     = nt * 16 + (lane & 15);
  const float bias = is_dec ? bq[n] : 0.0f;
  const int mbase = mt * 16 + ((lane >> 4) << 3);  // lanes 16-31 start at M=8
#pragma unroll
  for (int r = 0; r < 8; ++r) {
    Out[(size_t)(mbase + r) * H_ + n] = acc[r] + bias;
  }
}

// ---------------------------------------------------------------------------
// Phase 2: scores[b,d,e] = sum_h tanh(dec_f[b,d,h] + enc_f[b,e,h]) * v[h]
// 256-thread block computes a 16(d) x 16(e) tile; dec/enc rows staged in LDS.
// enc tile rows padded by 4 floats: row stride 260 dwords == 4 banks mod 64,
// so the 16 distinct enc rows read per cycle span all 64 LDS banks.
// Inner loop: ds_load_b128 x3 + 4x (v_tanh_f32 + v_fmac_f32) per 4 h-values.
// ---------------------------------------------------------------------------
__global__ void __launch_bounds__(256)
mlp_attn_score_tanh(const float* __restrict__ dec_f,
                    const float* __restrict__ enc_f,
                    const float* __restrict__ vvec,
                    float* __restrict__ out) {
  __shared__ float sdec[16][H_];        // 16 KB
  __shared__ float senc[16][H_ + 4];    // 16.25 KB (bank-conflict pad)
  __shared__ float sv[H_];              // 1 KB

  const int tid    = threadIdx.x;       // 0..255
  const int etile  = blockIdx.x;        // 0..31  -> e0 = etile*16
  const int bdtile = blockIdx.y;        // 0..63  -> 16 dec rows
  const int bd0    = bdtile * 16;       // global dec row base (b*DL + d)
  const int b      = bd0 >> 8;          // bd0 / DL_ ; uniform per block
  const int e0     = etile * 16;

  // Cooperative LDS fill (float4): 1024 float4 per tile, 4 per thread.
  const float4* __restrict__ dsrc =
      (const float4*)(dec_f + (size_t)bd0 * H_);
#pragma unroll
  for (int i = 0; i < 4; ++i) {
    int idx = tid + i * 256;                    // 0..1023
    ((float4*)&sdec[0][0])[idx] = dsrc[idx];    // dec tile is contiguous
  }
#pragma unroll
  for (int i = 0; i < 4; ++i) {
    int idx = tid + i * 256;                    // 0..1023
    int r   = idx >> 6;                         // 64 float4 per row
    int c4  = idx & 63;
    const float4* __restrict__ esrc =
        (const float4*)(enc_f + (size_t)(b * EL_ + e0 + r) * H_);
    *(float4*)&senc[r][c4 * 4] = esrc[c4];      // padded row keeps 16B align
  }
  sv[tid] = vvec[tid];                          // H_ == blockDim.x
  __syncthreads();

  const int dl = tid >> 4;
  const int el = tid & 15;

  float acc = 0.0f;
#pragma unroll 4
  for (int h = 0; h < H_; h += 4) {
    float4 dv = *(const float4*)&sdec[dl][h];
    float4 ev = *(const float4*)&senc[el][h];
    float4 vv = *(const float4*)&sv[h];
    acc = fmaf(TANH(dv.x + ev.x), vv.x, acc);
    acc = fmaf(TANH(dv.y + ev.y), vv.y, acc);
    acc = fmaf(TANH(dv.z + ev.z), vv.z, acc);
    acc = fmaf(TANH(dv.w + ev.w), vv.w, acc);
  }

  out[(size_t)(bd0 + dl) * EL_ + (e0 + el)] = acc;
}

extern "C" void kernel_launch(void* const* d_in, const int* in_sizes, int n_in,
                              void* d_out, int out_size, void* d_ws, size_t ws_size,
                              hipStream_t stream) {
  (void)in_sizes; (void)n_in; (void)out_size; (void)ws_size;

  const float* dec_in = (const float*)d_in[0];  // [B, DL, DDEC]
  const float* enc_in = (const float*)d_in[1];  // [B, EL, DENC]
  const float* Wq     = (const float*)d_in[2];  // [DDEC, H]
  const float* bq     = (const float*)d_in[3];  // [H]
  const float* Wc     = (const float*)d_in[4];  // [DENC, H]
  const float* vvec   = (const float*)d_in[5];  // [H]
  float*       out    = (float*)d_out;          // [B, DL, EL]

  // Workspace layout: dec_f [1024,256] f32 (1 MB) | enc_f [2048,256] f32 (2 MB)
  float* dec_f = (float*)d_ws;
  float* enc_f = dec_f + (size_t)(B_ * DL_) * H_;

  // Phase 1: 3072 tiles, 8 waves/block -> 384 blocks.
  mlp_attn_proj_wmma<<<384, 256, 0, stream>>>(dec_in, enc_in, Wq, bq, Wc,
                                              dec_f, enc_f);
  // Phase 2: grid = (EL/16, B*DL/16) = (32, 64), 256 threads.
  mlp_attn_score_tanh<<<dim3(32, 64), 256, 0, stream>>>(dec_f, enc_f, vvec, out);
}